// TransformerModel_32435593020281
// MI455X (gfx1250) — compile-verified
//
#include <hip/hip_runtime.h>
#include <math.h>

#define NNODES 50000
#define NEDGES 800000
#define HEADS  4
#define CH     64            // per-head channels
#define HC     (HEADS * CH)  // 256
#define LN_EPS 1e-5f

typedef float v2f __attribute__((ext_vector_type(2)));
typedef float v8f __attribute__((ext_vector_type(8)));

// ---------------------------------------------------------------- fills
__global__ void k_fill_f32(float* p, float v, int n) {
  int i = blockIdx.x * blockDim.x + threadIdx.x;
  if (i < n) p[i] = v;
}
__global__ void k_fill_u32(unsigned* p, unsigned v, int n) {
  int i = blockIdx.x * blockDim.x + threadIdx.x;
  if (i < n) p[i] = v;
}

// ---------------------------------------------------------------- WMMA GEMM
// out[M x Nc] = A[M x K] @ W[K x Nc] + bias   (row-major, f32, exact)
// One wave computes one 16x16 tile via 16x16x4 f32 WMMA, K/4 unrolled steps.
// K and Nc are compile-time so all loads/stores use constant immediate offsets.
// f32 A layout (ISA 7.12.2): lane L: M = L&15, VGPR0 = K=(L>>4)*2, VGPR1 = K+1.
// f32 B layout: lane L: N = L&15, VGPR0 = row K=(L>>4)*2, VGPR1 = row K+1.
// f32 C/D: VGPR r: lanes 0-15 -> M=r, lanes 16-31 -> M=r+8; N = lane&15.
template <int K, int Nc>
__global__ void k_gemm_wmma(const float* __restrict__ A,
                            const float* __restrict__ W,
                            const float* __restrict__ bias,
                            float* __restrict__ out, int M) {
  const int lane = threadIdx.x & 31;
  const int wid  = (blockIdx.x * blockDim.x + threadIdx.x) >> 5;
  constexpr int ntn = Nc / 16;
  const int total = (M >> 4) * ntn;
  if (wid >= total) return;                 // wave-uniform exit: EXEC stays all-1
  const int mt = wid / ntn;
  const int nt = wid - mt * ntn;
  const int lm = lane & 15;
  const int kb = (lane >> 4) << 1;          // 0 or 2
  const float* __restrict__ arow = A + (mt * 16 + lm) * K + kb;   // 8B aligned
  const float* __restrict__ wcol = W + kb * Nc + nt * 16 + lm;
  v8f acc = {};
#pragma unroll
  for (int k0 = 0; k0 < K; k0 += 4) {
    v2f a = *(const v2f*)(arow + k0);       // one global_load_b64, const offset
    v2f b;
    b.x = wcol[k0 * Nc];                    // const-offset global_load_b32
    b.y = wcol[k0 * Nc + Nc];
    acc = __builtin_amdgcn_wmma_f32_16x16x4_f32(false, a, false, b,
                                                (short)0, acc, false, false);
  }
  const int col = nt * 16 + lm;
  const float bv = bias[col];
  float* __restrict__ orow = out + (mt * 16 + ((lane >> 4) << 3)) * Nc + col;
#pragma unroll
  for (int r = 0; r < 8; ++r)
    orow[r * Nc] = acc[r] + bv;             // const-offset stores
}

// ---------------------------------------------------------------- edge pass A
// Wave per edge: alpha[e,h] = dot(q[dst,h,:], k[src,h,:]) / sqrt(C)
// 8 lanes per head, 8 channels per lane (two float4 loads per operand).
__global__ void k_edge_logits(const int* __restrict__ ei,
                              const float* __restrict__ q,
                              const float* __restrict__ k,
                              float* __restrict__ alpha,
                              unsigned* __restrict__ amaxU) {
  const int lane = threadIdx.x & 31;
  const int e = (blockIdx.x * blockDim.x + threadIdx.x) >> 5;
  if (e >= NEDGES) return;
  const int src = ei[e];
  const int dst = ei[NEDGES + e];
  const int h  = lane >> 3;          // 0..3
  const int c0 = (lane & 7) << 3;    // 0,8,...,56  (32B-aligned float4 pairs)
  const float4* qp = (const float4*)(q + (size_t)dst * HC + h * CH + c0);
  const float4* kp = (const float4*)(k + (size_t)src * HC + h * CH + c0);
  const float4 qa = qp[0], qb = qp[1];
  const float4 ka = kp[0], kb4 = kp[1];
  float s = qa.x * ka.x + qa.y * ka.y + qa.z * ka.z + qa.w * ka.w +
            qb.x * kb4.x + qb.y * kb4.y + qb.z * kb4.z + qb.w * kb4.w;
  s += __shfl_xor(s, 1, 8);
  s += __shfl_xor(s, 2, 8);
  s += __shfl_xor(s, 4, 8);
  if ((lane & 7) == 0) {
    float a = s * 0.125f;            // 1/sqrt(64)
    alpha[(size_t)e * HEADS + h] = a;
    unsigned bits = __float_as_uint(a);
    unsigned enc  = (bits & 0x80000000u) ? ~bits : (bits | 0x80000000u);
    atomicMax(&amaxU[(size_t)dst * HEADS + h], enc);
  }
}

// ---------------------------------------------------------------- edge pass B
// Thread per (edge, head): ex = exp(alpha - amax[dst]);  denom[dst] += ex
__global__ void k_edge_exp(const int* __restrict__ ei,
                           const unsigned* __restrict__ amaxU,
                           float* __restrict__ alpha,   // in: logits, out: ex
                           float* __restrict__ denom) {
  const long t = (long)blockIdx.x * blockDim.x + threadIdx.x;
  if (t >= (long)NEDGES * HEADS) return;
  const int e = (int)(t >> 2);
  const int h = (int)(t & 3);
  const int dst = ei[NEDGES + e];
  const unsigned u = amaxU[(size_t)dst * HEADS + h];
  const float amax = (u & 0x80000000u) ? __uint_as_float(u ^ 0x80000000u)
                                       : __uint_as_float(~u);
  const float ex = expf(alpha[t] - amax);
  alpha[t] = ex;
  atomicAdd(&denom[(size_t)dst * HEADS + h], ex);
}

// ---------------------------------------------------------------- edge pass C
// Wave per edge: acc[dst,c] += (1/H) * sum_h attn[e,h] * v[src,h,c]
__global__ void k_edge_aggr(const int* __restrict__ ei,
                            const float* __restrict__ ex,
                            const float* __restrict__ denom,
                            const float* __restrict__ v,
                            float* __restrict__ acc) {
  const int lane = threadIdx.x & 31;
  const int e = (blockIdx.x * blockDim.x + threadIdx.x) >> 5;
  if (e >= NEDGES) return;
  const int src = ei[e];
  const int dst = ei[NEDGES + e];
  float attn[HEADS];
#pragma unroll
  for (int h = 0; h < HEADS; ++h)
    attn[h] = ex[(size_t)e * HEADS + h] /
              (denom[(size_t)dst * HEADS + h] + 1e-16f);
  const float* vp = v + (size_t)src * HC;
  float* ap = acc + (size_t)dst * CH;
#pragma unroll
  for (int half = 0; half < 2; ++half) {
    const int c = lane + half * 32;
    float m = 0.f;
#pragma unroll
    for (int h = 0; h < HEADS; ++h) m += attn[h] * vp[h * CH + c];
    atomicAdd(&ap[c], m * (1.0f / HEADS));
  }
}

// ---------------------------------------------------------------- skip + LN (+relu)
// Wave per node over C=64 (2 elems/lane), shfl_xor reductions.
__global__ void k_skip_ln(const float* __restrict__ acc,
                          const float* __restrict__ s,
                          const float* __restrict__ g,
                          const float* __restrict__ b,
                          float* __restrict__ out, int relu) {
  const int lane = threadIdx.x & 31;
  const int n = (blockIdx.x * blockDim.x + threadIdx.x) >> 5;
  if (n >= NNODES) return;
  const size_t base = (size_t)n * CH;
  float x0 = acc[base + lane]      + s[base + lane];
  float x1 = acc[base + lane + 32] + s[base + lane + 32];
  float sum = x0 + x1;
#pragma unroll
  for (int off = 16; off >= 1; off >>= 1) sum += __shfl_xor(sum, off, 32);
  const float mu = sum * (1.f / CH);
  const float d0 = x0 - mu, d1 = x1 - mu;
  float vs = d0 * d0 + d1 * d1;
#pragma unroll
  for (int off = 16; off >= 1; off >>= 1) vs += __shfl_xor(vs, off, 32);
  const float rstd = rsqrtf(vs * (1.f / CH) + LN_EPS);
  float y0 = d0 * rstd * g[lane]      + b[lane];
  float y1 = d1 * rstd * g[lane + 32] + b[lane + 32];
  if (relu) { y0 = fmaxf(y0, 0.f); y1 = fmaxf(y1, 0.f); }
  out[base + lane]      = y0;
  out[base + lane + 32] = y1;
}

// ---------------------------------------------------------------- host
extern "C" void kernel_launch(void* const* d_in, const int* in_sizes, int n_in,
                              void* d_out, int out_size, void* d_ws, size_t ws_size,
                              hipStream_t stream) {
  const float* x   = (const float*)d_in[0];
  const int*   ei  = (const int*)  d_in[1];
  const float* Wq1 = (const float*)d_in[2],  *bq1 = (const float*)d_in[3];
  const float* Wk1 = (const float*)d_in[4],  *bk1 = (const float*)d_in[5];
  const float* Wv1 = (const float*)d_in[6],  *bv1 = (const float*)d_in[7];
  const float* Ws1 = (const float*)d_in[8],  *bs1 = (const float*)d_in[9];
  const float* Wq2 = (const float*)d_in[10], *bq2 = (const float*)d_in[11];
  const float* Wk2 = (const float*)d_in[12], *bk2 = (const float*)d_in[13];
  const float* Wv2 = (const float*)d_in[14], *bv2 = (const float*)d_in[15];
  const float* Ws2 = (const float*)d_in[16], *bs2 = (const float*)d_in[17];
  const float* ln_g = (const float*)d_in[18], *ln_b = (const float*)d_in[19];
  const float* Wf  = (const float*)d_in[20], *bf  = (const float*)d_in[21];

  // ---- workspace carve (floats)
  size_t off = 0;
  auto carve = [&](size_t nfloats) -> float* {
    float* p = (float*)((char*)d_ws + off);
    off += nfloats * sizeof(float);
    return p;
  };
  float*    q     = carve((size_t)NNODES * HC);
  float*    k     = carve((size_t)NNODES * HC);
  float*    v     = carve((size_t)NNODES * HC);
  float*    s     = carve((size_t)NNODES * CH);
  float*    alpha = carve((size_t)NEDGES * HEADS);
  float*    denom = carve((size_t)NNODES * HEADS);
  unsigned* amaxU = (unsigned*)carve((size_t)NNODES * HEADS);
  float*    acc   = carve((size_t)NNODES * CH);
  float*    h1    = carve((size_t)NNODES * CH);
  float*    h2    = carve((size_t)NNODES * CH);

  const dim3 blk(256);
  auto blocks_for_tiles = [](int M, int Nc) {
    return dim3(((M >> 4) * (Nc >> 4) + 7) / 8);   // 8 waves / 256-thread block
  };
  auto fillf = [&](float* p, float val, int n) {
    hipLaunchKernelGGL(k_fill_f32, dim3((n + 255) / 256), blk, 0, stream, p, val, n);
  };
  auto fillu = [&](unsigned* p, unsigned val, int n) {
    hipLaunchKernelGGL(k_fill_u32, dim3((n + 255) / 256), blk, 0, stream, p, val, n);
  };

  const int ewaves_blocks  = (NEDGES + 7) / 8;            // wave-per-edge kernels
  const int eh_blocks      = (NEDGES * HEADS + 255) / 256;
  const int node_wave_blks = (NNODES + 7) / 8;

  auto conv_layer = [&](const float* inNC,
                        const float* Wq, const float* bq, const float* Wk_,
                        const float* bk_, const float* Wv, const float* bv_,
                        const float* Ws_, const float* bs_,
                        float* outNC, int relu) {
    hipLaunchKernelGGL((k_gemm_wmma<64, 256>), blocks_for_tiles(NNODES, 256), blk,
                       0, stream, inNC, Wq,  bq,  q, NNODES);
    hipLaunchKernelGGL((k_gemm_wmma<64, 256>), blocks_for_tiles(NNODES, 256), blk,
                       0, stream, inNC, Wk_, bk_, k, NNODES);
    hipLaunchKernelGGL((k_gemm_wmma<64, 256>), blocks_for_tiles(NNODES, 256), blk,
                       0, stream, inNC, Wv,  bv_, v, NNODES);
    hipLaunchKernelGGL((k_gemm_wmma<64, 64>),  blocks_for_tiles(NNODES, 64),  blk,
                       0, stream, inNC, Ws_, bs_, s, NNODES);
    fillu(amaxU, 0u, NNODES * HEADS);          // enc(-inf)
    fillf(denom, 0.f, NNODES * HEADS);
    fillf(acc, 0.f, NNODES * CH);
    hipLaunchKernelGGL(k_edge_logits, dim3(ewaves_blocks), blk, 0, stream,
                       ei, q, k, alpha, amaxU);
    hipLaunchKernelGGL(k_edge_exp, dim3(eh_blocks), blk, 0, stream,
                       ei, amaxU, alpha, denom);
    hipLaunchKernelGGL(k_edge_aggr, dim3(ewaves_blocks), blk, 0, stream,
                       ei, alpha, denom, v, acc);
    hipLaunchKernelGGL(k_skip_ln, dim3(node_wave_blks), blk, 0, stream,
                       acc, s, ln_g, ln_b, outNC, relu);
  };

  // layer 1: x -> h1 (LN + ReLU)
  conv_layer(x, Wq1, bq1, Wk1, bk1, Wv1, bv1, Ws1, bs1, h1, 1);
  // layer 2: h1 -> h2 (LN, no ReLU)
  conv_layer(h1, Wq2, bq2, Wk2, bk2, Wv2, bv2, Ws2, bs2, h2, 0);
  // final: d_out = h2 @ Wf + bf   (50000 x 32)
  hipLaunchKernelGGL((k_gemm_wmma<64, 32>), blocks_for_tiles(NNODES, 32), blk,
                     0, stream, h2, Wf, bf, (float*)d_out, NNODES);
}